// FeatureOptimalLoss_57913339019376
// MI455X (gfx1250) — compile-verified
//
#include <hip/hip_runtime.h>
#include <hip/hip_bf16.h>

typedef __attribute__((ext_vector_type(2))) float v2f;
typedef __attribute__((ext_vector_type(8))) float v8f;

#define N 1024
#define DIM 256
#define EPS 100.0f
#define INV_EPS 0.01f
#define LOG_MU -6.93147180559945f   /* log(1/1024) */
#define NITER 10

#define HAVE_TDM   __has_builtin(__builtin_amdgcn_tensor_load_to_lds)
#define HAVE_ASYNC __has_builtin(__builtin_amdgcn_global_load_async_to_lds_b128)

// ---------------------------------------------------------------- init
__global__ void k_init(float* u, float* v, float* err, int* done, float* out) {
    int i = blockIdx.x * blockDim.x + threadIdx.x;
    if (i < N) { u[i] = 0.f; v[i] = 0.f; }
    if (i == 0) { *err = 0.f; *done = 0; *out = 0.f; }
}

// ------------------------------------------------- cost matrix C, K, K^T
// 32x32 output tile per 256-thread block, 2x2 per-thread register block,
// k staged through LDS in chunks of 64. LDS rows padded by 4 floats; the
// TDM path reproduces this padding in hardware via pad_interval/pad_amount.
#define TILE 32
#define KC 64

#if HAVE_TDM
typedef unsigned int u32x4 __attribute__((ext_vector_type(4)));
typedef int i32x8 __attribute__((ext_vector_type(8)));
typedef int i32x4 __attribute__((ext_vector_type(4)));

// 2D tile DMA: tile_w x tile_h elements (4B each) from a row-major tensor
// with row stride `stride` elements into LDS at byte offset lds_off,
// inserting 4 pad DWORDs after every 64 DWORDs (rows of 64 -> stride 68).
__device__ __forceinline__ void tdm_load_2d(const float* gsrc, unsigned lds_off,
                                            unsigned tile_w, unsigned tile_h,
                                            unsigned stride) {
    unsigned long long ga = (unsigned long long)(uintptr_t)gsrc;
    u32x4 g0;
    g0[0] = 1u;                                   // count=1 (valid), user mode
    g0[1] = lds_off;                              // lds_addr (bytes)
    g0[2] = (unsigned)ga;                         // global_addr[31:0]
    g0[3] = (unsigned)((ga >> 32) & 0x1FFFFFFu)   // global_addr[56:32]
            | 0x80000000u;                        // type=2 ("image")
    i32x8 g1;
    g1[0] = (int)((2u << 16)                      // data_size = 4 bytes
                | (1u << 20)                      // pad_enable
                | (5u << 22)                      // pad_interval: 64 DWORDs
                | (3u << 25));                    // pad_amount:   4 DWORDs
    g1[1] = (int)((stride & 0xFFFFu) << 16);      // tensor_dim0[15:0]
    g1[2] = (int)(((stride >> 16) & 0xFFFFu)      // tensor_dim0[31:16]
                | ((unsigned)(N & 0xFFFF) << 16));// tensor_dim1[15:0]
    g1[3] = (int)(((unsigned)N >> 16)             // tensor_dim1[31:16]
                | (tile_w << 16));                // tile_dim0
    g1[4] = (int)tile_h;                          // tile_dim1 (tile_dim2=0)
    g1[5] = (int)stride;                          // tensor_dim0_stride[31:0]
    g1[6] = 0;                                    // stride hi, dim1_stride lo
    g1[7] = 0;
    i32x4 gz4 = {0, 0, 0, 0};                     // groups 2/3 unused (2D)
    i32x8 gz8 = {0, 0, 0, 0, 0, 0, 0, 0};
    __builtin_amdgcn_tensor_load_to_lds(g0, g1, gz4, gz4, gz8, 0);
}
#endif

__global__ void k_cost(const float* __restrict__ X, const float* __restrict__ Y,
                       float* __restrict__ C, float* __restrict__ K,
                       float* __restrict__ KT) {
    __shared__ float xs[TILE][KC + 4];
    __shared__ float ys[TILE][KC + 4];
    const int t  = threadIdx.x;          // 0..255
    const int ti = t >> 4;               // 0..15
    const int tj = t & 15;               // 0..15
    const int ib = blockIdx.y * TILE;
    const int jb = blockIdx.x * TILE;

    float a00 = 0.f, a01 = 0.f, a10 = 0.f, a11 = 0.f;

    for (int kc = 0; kc < DIM; kc += KC) {
        __syncthreads();   // protect LDS reuse across chunks
#if HAVE_TDM
        if (threadIdx.x < 32) {          // wave 0 issues the tile DMAs
            tdm_load_2d(X + (size_t)ib * DIM + kc,
                        (unsigned)(uintptr_t)&xs[0][0], KC, TILE, DIM);
            tdm_load_2d(Y + (size_t)jb * DIM + kc,
                        (unsigned)(uintptr_t)&ys[0][0], KC, TILE, DIM);
            __builtin_amdgcn_s_wait_tensorcnt(0);
        }
        __syncthreads();
#elif HAVE_ASYNC
        for (int e = t; e < (TILE * KC) / 4; e += 256) {   // 512 float4 tiles
            int r = e >> 4, c = (e & 15) * 4;
            __builtin_amdgcn_global_load_async_to_lds_b128(
                (const __attribute__((address_space(1))) void*)
                    &X[(size_t)(ib + r) * DIM + kc + c],
                (__attribute__((address_space(3))) void*)&xs[r][c], 0, 0);
            __builtin_amdgcn_global_load_async_to_lds_b128(
                (const __attribute__((address_space(1))) void*)
                    &Y[(size_t)(jb + r) * DIM + kc + c],
                (__attribute__((address_space(3))) void*)&ys[r][c], 0, 0);
        }
        asm volatile("s_wait_asynccnt 0x0" ::: "memory");
        __syncthreads();
#else
        for (int e = t; e < TILE * KC; e += 256) {
            int r = e >> 6, c = e & 63;
            xs[r][c] = X[(size_t)(ib + r) * DIM + kc + c];
            ys[r][c] = Y[(size_t)(jb + r) * DIM + kc + c];
        }
        if (kc + KC < DIM) {
            int r = t >> 3, c = (t & 7) * 8;
            __builtin_prefetch(&X[(size_t)(ib + r) * DIM + kc + KC + c], 0, 0);
            __builtin_prefetch(&Y[(size_t)(jb + r) * DIM + kc + KC + c], 0, 0);
        }
        __syncthreads();
#endif

        const int i2 = ti * 2, j2 = tj * 2;
#pragma unroll 4
        for (int kk = 0; kk < KC; kk += 4) {
            float4 xa = *(const float4*)&xs[i2    ][kk];
            float4 xb = *(const float4*)&xs[i2 + 1][kk];
            float4 ya = *(const float4*)&ys[j2    ][kk];
            float4 yb = *(const float4*)&ys[j2 + 1][kk];
            a00 += fabsf(xa.x-ya.x)+fabsf(xa.y-ya.y)+fabsf(xa.z-ya.z)+fabsf(xa.w-ya.w);
            a01 += fabsf(xa.x-yb.x)+fabsf(xa.y-yb.y)+fabsf(xa.z-yb.z)+fabsf(xa.w-yb.w);
            a10 += fabsf(xb.x-ya.x)+fabsf(xb.y-ya.y)+fabsf(xb.z-ya.z)+fabsf(xb.w-ya.w);
            a11 += fabsf(xb.x-yb.x)+fabsf(xb.y-yb.y)+fabsf(xb.z-yb.z)+fabsf(xb.w-yb.w);
        }
    }

    const int i = ib + ti * 2, j = jb + tj * 2;
    C[(size_t)i * N + j]           = a00;
    C[(size_t)i * N + j + 1]       = a01;
    C[(size_t)(i + 1) * N + j]     = a10;
    C[(size_t)(i + 1) * N + j + 1] = a11;
    float k00 = expf(-a00 * INV_EPS), k01 = expf(-a01 * INV_EPS);
    float k10 = expf(-a10 * INV_EPS), k11 = expf(-a11 * INV_EPS);
    K[(size_t)i * N + j]           = k00;
    K[(size_t)i * N + j + 1]       = k01;
    K[(size_t)(i + 1) * N + j]     = k10;
    K[(size_t)(i + 1) * N + j + 1] = k11;
    KT[(size_t)j * N + i]           = k00;
    KT[(size_t)(j + 1) * N + i]     = k01;
    KT[(size_t)j * N + i + 1]       = k10;
    KT[(size_t)(j + 1) * N + i + 1] = k11;
}

// ---------------------------------------------- WMMA f32 mat-vec r = M w
// Block = 256 threads = 8 waves; block handles 16 output rows.
// Wave w covers columns [w*128, w*128+128) in K-steps of 4 via
// V_WMMA_F32_16X16X4_F32 with the vector broadcast into all 16 B columns.
__global__ void k_matvec_wmma(const float* __restrict__ M,
                              const float* __restrict__ w,
                              float* __restrict__ out) {
    __shared__ float red[8][16];
    const int wave = threadIdx.x >> 5;
    const int lane = threadIdx.x & 31;
    const int i0   = blockIdx.x * 16;
    const int row  = i0 + (lane & 15);
    const int kh   = lane >> 4;          // 0: k=0,1   1: k=2,3 (A layout)

    v8f acc = {};
    const int jbase = wave * 128;
#pragma unroll 4
    for (int jc = 0; jc < 128; jc += 4) {
        const int col = jbase + jc + kh * 2;
        v2f A = *(const v2f*)(M + (size_t)row * N + col);  // A[16x4] tile
        v2f B;                                             // B[4x16] bcast
        B.x = w[col];
        B.y = w[col + 1];
        acc = __builtin_amdgcn_wmma_f32_16x16x4_f32(
            false, A, false, B, (short)0, acc, false, false);
    }
    // D layout: VGPR q -> row (kh?8:0)+q, column = lane&15 (all cols equal)
    if ((lane & 15) == 0) {
#pragma unroll
        for (int q = 0; q < 8; ++q) red[wave][kh * 8 + q] = acc[q];
    }
    __syncthreads();
    if (threadIdx.x < 16) {
        float s = 0.f;
#pragma unroll
        for (int q = 0; q < 8; ++q) s += red[q][threadIdx.x];
        out[i0 + threadIdx.x] = s;
    }
}

// ----------------------------------------------------------- iteration
__global__ void k_expb(const float* v, float* b, float* err) {
    int i = blockIdx.x * blockDim.x + threadIdx.x;
    if (i < N) b[i] = expf(v[i] * INV_EPS);
    if (i == 0) *err = 0.f;
}

__global__ void k_update_u(const float* u, const float* rv, float* un,
                           float* a, float* err) {
    __shared__ float sm[256];
    int i = blockIdx.x * blockDim.x + threadIdx.x;
    float e = 0.f;
    if (i < N) {
        float uc   = u[i];
        float lse  = logf(expf(uc * INV_EPS) * rv[i] + 1e-6f);
        float unew = EPS * (LOG_MU - lse) + uc;
        un[i] = unew;
        a[i]  = expf(unew * INV_EPS);
        e     = fabsf(unew - uc);
    }
    sm[threadIdx.x] = e;
    __syncthreads();
    for (int o = 128; o > 0; o >>= 1) {
        if (threadIdx.x < o) sm[threadIdx.x] += sm[threadIdx.x + o];
        __syncthreads();
    }
    if (threadIdx.x == 0) atomicAdd(err, sm[0]);
}

__global__ void k_update_v(const float* v, const float* b, const float* sv,
                           float* vn) {
    int j = blockIdx.x * blockDim.x + threadIdx.x;
    if (j < N) {
        float lse = logf(b[j] * sv[j] + 1e-6f);
        vn[j] = EPS * (LOG_MU - lse) + v[j];
    }
}

__global__ void k_commit(float* u, float* v, const float* un, const float* vn,
                         const int* done) {
    int i = blockIdx.x * blockDim.x + threadIdx.x;
    if (i < N && *done == 0) { u[i] = un[i]; v[i] = vn[i]; }
}

__global__ void k_done(const float* err, int* done) {
    if (*err < 0.1f) *done = 1;
}

// ------------------------------------------------------------ epilogue
__global__ void k_ab(const float* u, const float* v, float* a, float* b) {
    int i = blockIdx.x * blockDim.x + threadIdx.x;
    if (i < N) { a[i] = expf(u[i] * INV_EPS); b[i] = expf(v[i] * INV_EPS); }
}

__global__ void k_loss(const float* __restrict__ C, const float* __restrict__ K,
                       const float* __restrict__ a, const float* __restrict__ b,
                       float* out) {
    __shared__ float sm[256];
    size_t gid = (size_t)blockIdx.x * blockDim.x + threadIdx.x;
    float p = 0.f;
    for (size_t e = gid; e < (size_t)N * N; e += (size_t)256 * 256) {
        int i = (int)(e >> 10), j = (int)(e & (N - 1));
        p += K[e] * C[e] * a[i] * b[j];
    }
    sm[threadIdx.x] = p;
    __syncthreads();
    for (int o = 128; o > 0; o >>= 1) {
        if (threadIdx.x < o) sm[threadIdx.x] += sm[threadIdx.x + o];
        __syncthreads();
    }
    if (threadIdx.x == 0) atomicAdd(out, sm[0]);
}

// ---------------------------------------------------------------- host
extern "C" void kernel_launch(void* const* d_in, const int* in_sizes, int n_in,
                              void* d_out, int out_size, void* d_ws, size_t ws_size,
                              hipStream_t stream) {
    const float* X = (const float*)d_in[0];
    const float* Y = (const float*)d_in[1];
    float* out = (float*)d_out;
    float* ws  = (float*)d_ws;

    const size_t M2 = (size_t)N * N;
    float* C  = ws;
    float* K  = ws + M2;
    float* KT = ws + 2 * M2;
    float* base = ws + 3 * M2;
    float* u  = base;
    float* v  = base + N;
    float* un = base + 2 * N;
    float* vn = base + 3 * N;
    float* a  = base + 4 * N;
    float* b  = base + 5 * N;
    float* r  = base + 6 * N;
    float* s  = base + 7 * N;
    float* err = base + 8 * N;
    int*   done = (int*)(base + 8 * N + 1);

    k_init<<<dim3(N / 256), dim3(256), 0, stream>>>(u, v, err, done, out);
    k_cost<<<dim3(N / TILE, N / TILE), dim3(256), 0, stream>>>(X, Y, C, K, KT);

    for (int it = 0; it < NITER; ++it) {
        k_expb<<<dim3(N / 256), dim3(256), 0, stream>>>(v, b, err);
        k_matvec_wmma<<<dim3(N / 16), dim3(256), 0, stream>>>(K, b, r);
        k_update_u<<<dim3(N / 256), dim3(256), 0, stream>>>(u, r, un, a, err);
        k_matvec_wmma<<<dim3(N / 16), dim3(256), 0, stream>>>(KT, a, s);
        k_update_v<<<dim3(N / 256), dim3(256), 0, stream>>>(v, b, s, vn);
        k_commit<<<dim3(N / 256), dim3(256), 0, stream>>>(u, v, un, vn, done);
        k_done<<<dim3(1), dim3(1), 0, stream>>>(err, done);
    }

    k_ab<<<dim3(N / 256), dim3(256), 0, stream>>>(u, v, a, b);
    k_loss<<<dim3(256), dim3(256), 0, stream>>>(C, K, a, b, out);
}